// LambdaLoss_79852031967377
// MI455X (gfx1250) — compile-verified
//
#include <hip/hip_runtime.h>
#include <stdint.h>

// LambdaLoss (LambdaRank gradient) for a=[32,1024] f32 scores, y=[32,1024] f32 labels in {0..4}.
// Output: d_out[0] = sum(a); d_out[1..32768] = grad[b,n].
//
// Compute-bound pairwise op (no GEMM structure) -> optimize for VALU/transcendental
// throughput; gfx1250 async global->LDS (B128) staging + s_wait_asynccnt for row tiles.

#define N       1024
#define CHUNK   128
#define EPSF    1.1920929e-07f
#define SIGMAF  1.0f
#define LOG2E_F 1.44269504f

// Low 32 bits of a generic pointer into the shared aperture == LDS byte offset
// used as the VDST LDS-address VGPR for async loads.
__device__ __forceinline__ unsigned lds_off32(const void* p) {
  return (unsigned)(uintptr_t)p;
}

// CDNA5 async copy: 16 bytes global -> LDS, tracked by ASYNCcnt (no VGPR data path).
__device__ __forceinline__ void async_ld_b128(unsigned lds_byte_off, const float* g) {
  unsigned long long ga = (unsigned long long)(uintptr_t)g;
  asm volatile("global_load_async_to_lds_b128 %0, %1, off"
               :: "v"(lds_byte_off), "v"(ga)
               : "memory");
}

__device__ __forceinline__ void wait_async0() {
  asm volatile("s_wait_asynccnt 0" ::: "memory");
}

// Stage one 1024-float row into LDS: 128 lanes x 16B x 2 passes, fully unrolled.
__device__ __forceinline__ void stage_row(float* lds_dst, const float* grow, int t) {
#pragma unroll
  for (int u = 0; u < (N / (CHUNK * 4)); ++u) {
    const int k = (u * CHUNK + t) * 4;
    async_ld_b128(lds_off32(&lds_dst[k]), grow + k);
  }
}

// ---------------------------------------------------------------------------
// Kernel 1: per-row ranks -> il = 1/log2(rank+1), py = 2^y; chunk 0 also
// computes IDCG (via label histogram, labels are ints 0..4) and the row sum.
// Grid: (b, N/CHUNK), block: CHUNK.
// ---------------------------------------------------------------------------
__global__ void __launch_bounds__(CHUNK)
lambda_stats_kernel(const float* __restrict__ a, const float* __restrict__ y,
                    float* __restrict__ ws_il, float* __restrict__ ws_py,
                    float* __restrict__ ws_invidcg, float* __restrict__ ws_rowsum) {
  __shared__ __align__(16) float sa[N];
  __shared__ float redf[CHUNK];
  __shared__ unsigned long long red64[CHUNK];

  const int b = blockIdx.x;
  const int chunk = blockIdx.y;
  const int t = threadIdx.x;
  const float* arow = a + b * N;
  const float* yrow = y + b * N;

  stage_row(sa, arow, t);
  wait_async0();
  __syncthreads();

  // rank_i = 1 + #{ j : a_j > a_i  or (a_j == a_i and j < i) }  (stable argsort tiebreak)
  const int i = chunk * CHUNK + t;
  const float ai = sa[i];
  int cnt = 0;
#pragma unroll 8
  for (int j = 0; j < N; ++j) {
    float aj = sa[j];
    cnt += (int)(aj > ai) | ((int)(aj == ai) & (int)(j < i));
  }
  const int rank = 1 + cnt;

  const float il = __builtin_amdgcn_rcpf(__builtin_amdgcn_logf((float)(rank + 1)));
  const float yi = yrow[i];
  ws_il[b * N + i] = il;
  ws_py[b * N + i] = __builtin_amdgcn_exp2f(yi);   // exact for y in {0..4}

  if (chunk == 0) {
    // Histogram of labels 1..4 packed 16 bits each in a u64 (counts <= 1024),
    // plus the row sum of a for the placeholder loss.
    unsigned long long pk = 0ull;
    float rs = 0.0f;
#pragma unroll
    for (int u = 0; u < N / CHUNK; ++u) {
      const int k = u * CHUNK + t;
      int v = (int)yrow[k];
      if (v > 0) pk += 1ull << (16 * (v - 1));
      rs += sa[k];
    }
    red64[t] = pk;
    redf[t] = rs;
    __syncthreads();
    for (int s = CHUNK / 2; s > 0; s >>= 1) {
      if (t < s) { red64[t] += red64[t + s]; redf[t] += redf[t + s]; }
      __syncthreads();
    }
    const unsigned long long tot = red64[0];
    const float rowsum = redf[0];
    __syncthreads();

    const int c1 = (int)(tot & 0xFFFFu);
    const int c2 = (int)((tot >> 16) & 0xFFFFu);
    const int c3 = (int)((tot >> 32) & 0xFFFFu);
    const int c4 = (int)((tot >> 48) & 0xFFFFu);
    const int cum4 = c4;
    const int cum3 = cum4 + c3;
    const int cum2 = cum3 + c2;
    const int cum1 = cum2 + c1;

    // IDCG = sum over sorted positions p (0-based): (2^v - 1) / log2(p + 2)
    float part = 0.0f;
#pragma unroll
    for (int u = 0; u < N / CHUNK; ++u) {
      const int p = u * CHUNK + t;
      int v = (p < cum4) ? 4 : (p < cum3) ? 3 : (p < cum2) ? 2 : (p < cum1) ? 1 : 0;
      float gain = (float)((1 << v) - 1);
      part += gain * __builtin_amdgcn_rcpf(__builtin_amdgcn_logf((float)(p + 2)));
    }
    redf[t] = part;
    __syncthreads();
    for (int s = CHUNK / 2; s > 0; s >>= 1) {
      if (t < s) redf[t] += redf[t + s];
      __syncthreads();
    }
    if (t == 0) {
      ws_invidcg[b] = __builtin_amdgcn_rcpf(redf[0] + EPSF);
      ws_rowsum[b] = rowsum;
    }
  }
}

// ---------------------------------------------------------------------------
// Kernel 2: grad_i = -2*sigma*invIDCG * sum_j sgn_ij*|py_i-py_j|*|il_i-il_j| / (1+exp(sgn_ij*(a_i-a_j)))
// sgn folded as copysign: valid because the weight is exactly 0 when y_i == y_j.
// Grid: (b, N/CHUNK), block: CHUNK. 12 KB LDS, conflict-free broadcast reads.
// ---------------------------------------------------------------------------
__global__ void __launch_bounds__(CHUNK)
lambda_grad_kernel(const float* __restrict__ a,
                   const float* __restrict__ ws_il, const float* __restrict__ ws_py,
                   const float* __restrict__ ws_invidcg, const float* __restrict__ ws_rowsum,
                   float* __restrict__ out, int batches) {
  __shared__ __align__(16) float sa[N];
  __shared__ __align__(16) float spy[N];
  __shared__ __align__(16) float sil[N];

  const int b = blockIdx.x;
  const int chunk = blockIdx.y;
  const int t = threadIdx.x;

  stage_row(sa,  a     + b * N, t);
  stage_row(spy, ws_py + b * N, t);
  stage_row(sil, ws_il + b * N, t);
  wait_async0();
  __syncthreads();

  const int i = chunk * CHUNK + t;
  const float ai  = sa[i];
  const float pyi = spy[i];
  const float ili = sil[i];
  const float invidcg = ws_invidcg[b];

  float acc = 0.0f;
#pragma unroll 8
  for (int j = 0; j < N; ++j) {
    float aj = sa[j];
    float dy = pyi - spy[j];                       // sign(dy) == sign(y_i - y_j)
    float w  = fabsf(dy) * fabsf(ili - sil[j]);    // |delta_ndcg| * idcg (factored); w >= 0
    float x  = (ai - aj) * __builtin_copysignf(LOG2E_F, dy);  // sgn*(a_i-a_j)*log2(e)
    float e  = __builtin_amdgcn_exp2f(x);                     // exp(sgn * (a_i - a_j))
    acc = fmaf(__builtin_copysignf(w, dy),                    // sgn * w  (w >= 0)
               __builtin_amdgcn_rcpf(1.0f + e), acc);
  }
  out[1 + b * N + i] = -2.0f * SIGMAF * invidcg * acc;

  // Placeholder loss = sum(a); row sums were produced by kernel 1.
  if (b == 0 && chunk == 0 && t == 0) {
    float ls = 0.0f;
    for (int k = 0; k < batches; ++k) ls += ws_rowsum[k];
    out[0] = ls;
  }
}

// ---------------------------------------------------------------------------
extern "C" void kernel_launch(void* const* d_in, const int* in_sizes, int n_in,
                              void* d_out, int out_size, void* d_ws, size_t ws_size,
                              hipStream_t stream) {
  const float* a = (const float*)d_in[0];
  const float* y = (const float*)d_in[1];
  float* out = (float*)d_out;

  const int total = in_sizes[0];
  const int b = total / N;   // 32 for the reference shapes

  float* ws_il      = (float*)d_ws;                 // b*N
  float* ws_py      = ws_il + (size_t)b * N;        // b*N
  float* ws_invidcg = ws_py + (size_t)b * N;        // b
  float* ws_rowsum  = ws_invidcg + b;               // b

  dim3 grid((unsigned)b, N / CHUNK);
  lambda_stats_kernel<<<grid, CHUNK, 0, stream>>>(a, y, ws_il, ws_py,
                                                  ws_invidcg, ws_rowsum);
  lambda_grad_kernel<<<grid, CHUNK, 0, stream>>>(a, ws_il, ws_py,
                                                 ws_invidcg, ws_rowsum, out, b);
}